// ManoHand_45354854646000
// MI455X (gfx1250) — compile-verified
//
#include <hip/hip_runtime.h>
#include <hip/hip_bf16.h>

// ---------------- MANO constants ----------------
#define BATCH 8192
#define NVERT 778
#define NJNT  16
#define NSHP  10
#define NPOSE 135
#define NPOSEP 136        // padded K for WMMA (pf[135] = 0)
#define NCOL  (NVERT*3)   // 2334
#define PPC   12          // pose PCA params

typedef __attribute__((ext_vector_type(2))) float v2f;
typedef __attribute__((ext_vector_type(8))) float v8f;
typedef __attribute__((address_space(3))) float lds_f32;

// D = A(16x4 f32) * B(4x16 f32) + C   -- CDNA5 wave32 WMMA
#define WMMA4(a, b, c) \
  __builtin_amdgcn_wmma_f32_16x16x4_f32(false, (a), false, (b), (short)0, (c), false, false)

// ---------------------------------------------------------------------------
// Kernel 0: precompute SJ[k][j][c] = sum_v shapedirs[k, v*3+c] * Jreg[v][j]
//           Jt[j][c]    = sum_v v_template[v][c]  * Jreg[v][j]
// so that per-batch joints J[b,j,c] = Jt[j,c] + sum_k beta[b,k]*SJ[k,j,c]
// ---------------------------------------------------------------------------
__global__ void mano_precompute(const float* __restrict__ shapedirs,
                                const float* __restrict__ v_template,
                                const float* __restrict__ Jreg,
                                float* __restrict__ SJ, float* __restrict__ Jt)
{
  int idx = blockIdx.x * 256 + threadIdx.x;
  if (idx < NSHP * 48) {
    int k = idx / 48, r = idx % 48, j = r / 3, c = r % 3;
    float s = 0.f;
    for (int v = 0; v < NVERT; v++)
      s += shapedirs[(size_t)k * NCOL + v * 3 + c] * Jreg[v * NJNT + j];
    SJ[idx] = s;
  } else if (idx < NSHP * 48 + 48) {
    int r = idx - NSHP * 48, j = r / 3, c = r % 3;
    float s = 0.f;
    for (int v = 0; v < NVERT; v++)
      s += v_template[v * 3 + c] * Jreg[v * NJNT + j];
    Jt[r] = s;
  }
}

// ---------------------------------------------------------------------------
// Kernel A: per-batch pose pipeline (1 thread / batch element)
// ---------------------------------------------------------------------------
__device__ __forceinline__ void rodrigues3(const float aa[3], float R[9])
{
  float a0 = aa[0] + 1e-8f, a1 = aa[1] + 1e-8f, a2 = aa[2] + 1e-8f;
  float ang = sqrtf(a0 * a0 + a1 * a1 + a2 * a2);
  float inva = 1.0f / ang;
  float s, c;
  __sincosf(0.5f * ang, &s, &c);
  float qw = c, qx = s * aa[0] * inva, qy = s * aa[1] * inva, qz = s * aa[2] * inva;
  float rn = rsqrtf(qw * qw + qx * qx + qy * qy + qz * qz);
  qw *= rn; qx *= rn; qy *= rn; qz *= rn;
  float w2 = qw * qw, x2 = qx * qx, y2 = qy * qy, z2 = qz * qz;
  float wx = qw * qx, wy = qw * qy, wz = qw * qz;
  float xy = qx * qy, xz = qx * qz, yz = qy * qz;
  R[0] = w2 + x2 - y2 - z2; R[1] = 2.f * (xy - wz);     R[2] = 2.f * (wy + xz);
  R[3] = 2.f * (wz + xy);   R[4] = w2 - x2 + y2 - z2;   R[5] = 2.f * (yz - wx);
  R[6] = 2.f * (xz - wy);   R[7] = 2.f * (wx + yz);     R[8] = w2 - x2 - y2 + z2;
}

__global__ void __launch_bounds__(256) mano_pose(
    const float* __restrict__ beta, const float* __restrict__ theta,
    const float* __restrict__ hands_mean, const float* __restrict__ hc,
    const float* __restrict__ SJ, const float* __restrict__ Jt,
    float* __restrict__ pf, float* __restrict__ Amat)
{
  __shared__ float hc_s[PPC * 45];
  __shared__ float hm_s[45];
  __shared__ float SJ_s[NSHP * 48];
  __shared__ float Jt_s[48];
  int tid = threadIdx.x;
  for (int i = tid; i < PPC * 45; i += 256) hc_s[i] = hc[i];          // rows 0..11
  for (int i = tid; i < 45; i += 256) hm_s[i] = hands_mean[i];
  for (int i = tid; i < NSHP * 48; i += 256) SJ_s[i] = SJ[i];
  for (int i = tid; i < 48; i += 256) Jt_s[i] = Jt[i];
  __syncthreads();

  int b = blockIdx.x * 256 + tid;   // grid = 32 blocks -> exactly 8192

  float bt[NSHP];
  for (int k = 0; k < NSHP; k++) bt[k] = beta[(size_t)b * NSHP + k];

  float J[NJNT][3];
  for (int j = 0; j < NJNT; j++)
    for (int c = 0; c < 3; c++) {
      float s = Jt_s[j * 3 + c];
      for (int k = 0; k < NSHP; k++) s += bt[k] * SJ_s[k * 48 + j * 3 + c];
      J[j][c] = s;
    }

  float th[PPC];
  for (int k = 0; k < PPC; k++) th[k] = theta[(size_t)b * 15 + 3 + k];
  float fl[45];
  for (int q = 0; q < 45; q++) {
    float s = hm_s[q];
    for (int k = 0; k < PPC; k++) s += th[k] * hc_s[k * 45 + q];
    fl[q] = s;
  }

  float G0[12], Gp[12];
  for (int i = 0; i < NJNT; i++) {
    float aa[3];
    if (i == 0) { aa[0] = theta[(size_t)b * 15 + 0]; aa[1] = theta[(size_t)b * 15 + 1]; aa[2] = theta[(size_t)b * 15 + 2]; }
    else        { aa[0] = fl[(i - 1) * 3 + 0]; aa[1] = fl[(i - 1) * 3 + 1]; aa[2] = fl[(i - 1) * 3 + 2]; }
    float R[9];
    rodrigues3(aa, R);
    if (i >= 1) {
      size_t base = (size_t)b * NPOSEP + (i - 1) * 9;
      for (int e = 0; e < 9; e++)
        pf[base + e] = R[e] - ((e == 0 || e == 4 || e == 8) ? 1.f : 0.f);
    }
    float Gn[12];
    if (i == 0) {
      for (int r = 0; r < 3; r++) {
        Gn[r * 4 + 0] = R[r * 3 + 0]; Gn[r * 4 + 1] = R[r * 3 + 1];
        Gn[r * 4 + 2] = R[r * 3 + 2]; Gn[r * 4 + 3] = J[0][r];
      }
    } else {
      int p = ((i - 1) % 3 == 0) ? 0 : i - 1;          // PARENTS table
      float t0 = J[i][0] - J[p][0], t1 = J[i][1] - J[p][1], t2 = J[i][2] - J[p][2];
      const float* Gq = ((i - 1) % 3 == 0) ? G0 : Gp;
      for (int r = 0; r < 3; r++) {
        for (int c = 0; c < 3; c++)
          Gn[r * 4 + c] = Gq[r * 4 + 0] * R[0 * 3 + c] + Gq[r * 4 + 1] * R[1 * 3 + c] + Gq[r * 4 + 2] * R[2 * 3 + c];
        Gn[r * 4 + 3] = Gq[r * 4 + 0] * t0 + Gq[r * 4 + 1] * t1 + Gq[r * 4 + 2] * t2 + Gq[r * 4 + 3];
      }
    }
    // A' = G with translation column replaced by t - R_glob * J_i
    size_t ab = (size_t)b * 256 + i * 16;
    for (int r = 0; r < 3; r++) {
      Amat[ab + r * 4 + 0] = Gn[r * 4 + 0];
      Amat[ab + r * 4 + 1] = Gn[r * 4 + 1];
      Amat[ab + r * 4 + 2] = Gn[r * 4 + 2];
      Amat[ab + r * 4 + 3] = Gn[r * 4 + 3] -
          (Gn[r * 4 + 0] * J[i][0] + Gn[r * 4 + 1] * J[i][1] + Gn[r * 4 + 2] * J[i][2]);
    }
    Amat[ab + 12] = 0.f; Amat[ab + 13] = 0.f; Amat[ab + 14] = 0.f; Amat[ab + 15] = 1.f;
    if (i == 0) for (int e = 0; e < 12; e++) G0[e] = Gn[e];
    for (int e = 0; e < 12; e++) Gp[e] = Gn[e];
  }
  pf[(size_t)b * NPOSEP + NPOSE] = 0.f;   // K padding
}

// ---------------------------------------------------------------------------
// Kernel B: fused blendshape GEMM + skinning + joint regression.
// Workgroup = 16 batches x all 778 vertices, 4 waves round-robin vertex tiles.
// Branchless: all OOB handling by address clamping + zero-padded A operands.
// ---------------------------------------------------------------------------
__global__ void __launch_bounds__(128, 1) mano_lbs(
    const float* __restrict__ posedirs, const float* __restrict__ shapedirs,
    const float* __restrict__ v_template, const float* __restrict__ weights,
    const float* __restrict__ Jreg, const float* __restrict__ beta,
    const float* __restrict__ pf, const float* __restrict__ Amat,
    float* __restrict__ out)
{
  __shared__ float pf_s[16 * NPOSEP];       // pose features, 16 batches
  __shared__ float beta_s[16 * 12];         // beta padded to K=12
  __shared__ float A_s[16 * NJNT * 12];     // skinning mats, 3x4 per joint
  __shared__ float trans_s[4][16 * 17];     // per-wave verts transpose staging
  __shared__ float jred[4][3][16 * 16];     // per-wave joint partials

  const int tid = threadIdx.x;
  const int wid = tid >> 5;
  const int lane = tid & 31;
  const int m = lane & 15;                  // M row (batch) for A op / N col for B op
  const int hi = lane >> 4;                 // K-half selector (wave32 WMMA layout)
  const int b0 = blockIdx.x * 16;

  __builtin_prefetch(posedirs, 0, 3);       // global_prefetch_b8

  // ---- stage skinning matrices via CDNA5 async global->LDS DMA ----
  for (int i = tid; i < 16 * NJNT * 12; i += 128) {
    int bb = i / (NJNT * 12), r = i % (NJNT * 12);
    int j = r / 12, q = r % 12;
    const float* src = Amat + (size_t)(b0 + bb) * 256 + j * 16 + q;
    lds_f32* dst = (lds_f32*)&A_s[i];
    asm volatile("global_load_async_to_lds_b32 %0, %1, off"
                 :: "v"(dst), "v"(src) : "memory");
  }
  // ---- regular cooperative staging ----
  for (int i = tid; i < 16 * NPOSEP; i += 128)
    pf_s[i] = pf[(size_t)(b0 + i / NPOSEP) * NPOSEP + (i % NPOSEP)];
  for (int i = tid; i < 16 * 12; i += 128) {
    int bb = i / 12, k = i % 12;
    beta_s[i] = (k < NSHP) ? beta[(size_t)(b0 + bb) * NSHP + k] : 0.f;
  }
  asm volatile("s_wait_asynccnt 0" ::: "memory");
  __syncthreads();

  v8f jax = {}, jay = {}, jaz = {};

  for (int t = wid; t < 49; t += 4) {       // 49 vertex tiles of 16 (778 padded)
    const int v0 = t * 16;
    const int vv = v0 + m;
    const bool vok = vv < NVERT;
    const int vvc = vok ? vv : (NVERT - 1); // clamped vertex (branchless OOB)

    const float* sd = shapedirs + (size_t)vvc * 3;
    const float* pd = posedirs + (size_t)vvc * 3;

    // ---- v_posed tile [M=batch, N=vertex] for x/y/z via WMMA K-chains ----
    v8f acc0 = {}, acc1 = {}, acc2 = {};
#pragma unroll
    for (int ks = 0; ks < 12; ks += 4) {    // shapedirs, K padded 10->12
      const int ka = ks + 2 * hi;
      const int k0 = (ka < NSHP) ? ka : (NSHP - 1);       // A side is 0 there
      const int k1 = (ka + 1 < NSHP) ? (ka + 1) : (NSHP - 1);
      v2f a; a.x = beta_s[m * 12 + ka]; a.y = beta_s[m * 12 + ka + 1];
      v2f b0v, b1v, b2v;
      b0v.x = sd[(size_t)k0 * NCOL + 0]; b0v.y = sd[(size_t)k1 * NCOL + 0];
      b1v.x = sd[(size_t)k0 * NCOL + 1]; b1v.y = sd[(size_t)k1 * NCOL + 1];
      b2v.x = sd[(size_t)k0 * NCOL + 2]; b2v.y = sd[(size_t)k1 * NCOL + 2];
      acc0 = WMMA4(a, b0v, acc0);
      acc1 = WMMA4(a, b1v, acc1);
      acc2 = WMMA4(a, b2v, acc2);
    }
#pragma unroll
    for (int ks = 0; ks < NPOSEP; ks += 4) { // posedirs, K padded 135->136
      const int ka = ks + 2 * hi;
      const int k0 = (ka < NPOSE) ? ka : (NPOSE - 1);
      const int k1 = (ka + 1 < NPOSE) ? (ka + 1) : (NPOSE - 1);
      v2f a; a.x = pf_s[m * NPOSEP + ka]; a.y = pf_s[m * NPOSEP + ka + 1];
      v2f b0v, b1v, b2v;
      b0v.x = pd[(size_t)k0 * NCOL + 0]; b0v.y = pd[(size_t)k1 * NCOL + 0];
      b1v.x = pd[(size_t)k0 * NCOL + 1]; b1v.y = pd[(size_t)k1 * NCOL + 1];
      b2v.x = pd[(size_t)k0 * NCOL + 2]; b2v.y = pd[(size_t)k1 * NCOL + 2];
      acc0 = WMMA4(a, b0v, acc0);
      acc1 = WMMA4(a, b1v, acc1);
      acc2 = WMMA4(a, b2v, acc2);
    }
    {
      float vt0 = v_template[vvc * 3 + 0];
      float vt1 = v_template[vvc * 3 + 1];
      float vt2 = v_template[vvc * 3 + 2];
#pragma unroll
      for (int q = 0; q < 8; q++) { acc0[q] += vt0; acc1[q] += vt1; acc2[q] += vt2; }
    }

    // ---- skinning: verts = sum_j W[v,j] * (A'[b,j] . [v_posed,1]) ----
    // outer j rolled (weights streamed from L2), inner q unrolled (const idx)
    v8f vx = {}, vy = {}, vz = {};
#pragma unroll 2
    for (int j = 0; j < NJNT; j++) {
      float wload = weights[(size_t)vvc * NJNT + j];
      float wj = vok ? wload : 0.f;         // single cndmask zeroes OOB lanes
#pragma unroll
      for (int q = 0; q < 8; q++) {
        const float* Ab = &A_s[(q + 8 * hi) * (NJNT * 12) + j * 12];
        const float4 r0 = *(const float4*)(Ab + 0);
        const float4 r1 = *(const float4*)(Ab + 4);
        const float4 r2 = *(const float4*)(Ab + 8);
        float px = acc0[q], py = acc1[q], pz = acc2[q];
        float d0 = r0.x * px + r0.y * py + r0.z * pz + r0.w;
        float d1 = r1.x * px + r1.y * py + r1.z * pz + r1.w;
        float d2 = r2.x * px + r2.y * py + r2.z * pz + r2.w;
        vx[q] = fmaf(wj, d0, vx[q]);
        vy[q] = fmaf(wj, d1, vy[q]);
        vz[q] = fmaf(wj, d2, vz[q]);
      }
    }

    // ---- finger vertices (FINGER_IDX = [734,333,443,555,678]) ----
    int fidx = -1;
    if      (vv == 734) fidx = 0;
    else if (vv == 333) fidx = 1;
    else if (vv == 443) fidx = 2;
    else if (vv == 555) fidx = 3;
    else if (vv == 678) fidx = 4;
    if (fidx >= 0) {
#pragma unroll
      for (int q = 0; q < 8; q++) {
        size_t ob = (size_t)(b0 + q + 8 * hi) * 63 + (size_t)(16 + fidx) * 3;
        out[ob + 0] = vx[q]; out[ob + 1] = vy[q]; out[ob + 2] = vz[q];
      }
    }

    // ---- joints partials: joints_c[b,j] += verts_c[b,:] . Jreg[:,j] ----
    // verts transposed through per-wave LDS slab into WMMA A-operand layout.
    {
      float* tr = &trans_s[wid][0];
#pragma unroll
      for (int q = 0; q < 8; q++) tr[(q + 8 * hi) * 17 + m] = vx[q];
      asm volatile("s_wait_dscnt 0" ::: "memory");
#pragma unroll
      for (int kv = 0; kv < 16; kv += 4) {
        const int ka = kv + 2 * hi;
        const int i0 = (v0 + ka < NVERT) ? (v0 + ka) : (NVERT - 1);
        const int i1 = (v0 + ka + 1 < NVERT) ? (v0 + ka + 1) : (NVERT - 1);
        v2f a; a.x = tr[m * 17 + ka]; a.y = tr[m * 17 + ka + 1];
        v2f bb; bb.x = Jreg[(size_t)i0 * NJNT + m]; bb.y = Jreg[(size_t)i1 * NJNT + m];
        jax = WMMA4(a, bb, jax);
      }
#pragma unroll
      for (int q = 0; q < 8; q++) tr[(q + 8 * hi) * 17 + m] = vy[q];
      asm volatile("s_wait_dscnt 0" ::: "memory");
#pragma unroll
      for (int kv = 0; kv < 16; kv += 4) {
        const int ka = kv + 2 * hi;
        const int i0 = (v0 + ka < NVERT) ? (v0 + ka) : (NVERT - 1);
        const int i1 = (v0 + ka + 1 < NVERT) ? (v0 + ka + 1) : (NVERT - 1);
        v2f a; a.x = tr[m * 17 + ka]; a.y = tr[m * 17 + ka + 1];
        v2f bb; bb.x = Jreg[(size_t)i0 * NJNT + m]; bb.y = Jreg[(size_t)i1 * NJNT + m];
        jay = WMMA4(a, bb, jay);
      }
#pragma unroll
      for (int q = 0; q < 8; q++) tr[(q + 8 * hi) * 17 + m] = vz[q];
      asm volatile("s_wait_dscnt 0" ::: "memory");
#pragma unroll
      for (int kv = 0; kv < 16; kv += 4) {
        const int ka = kv + 2 * hi;
        const int i0 = (v0 + ka < NVERT) ? (v0 + ka) : (NVERT - 1);
        const int i1 = (v0 + ka + 1 < NVERT) ? (v0 + ka + 1) : (NVERT - 1);
        v2f a; a.x = tr[m * 17 + ka]; a.y = tr[m * 17 + ka + 1];
        v2f bb; bb.x = Jreg[(size_t)i0 * NJNT + m]; bb.y = Jreg[(size_t)i1 * NJNT + m];
        jaz = WMMA4(a, bb, jaz);
      }
    }
  }

  // ---- cross-wave reduction of joint accumulators ----
#pragma unroll
  for (int q = 0; q < 8; q++) {
    int bb = q + 8 * hi;
    jred[wid][0][bb * 16 + m] = jax[q];
    jred[wid][1][bb * 16 + m] = jay[q];
    jred[wid][2][bb * 16 + m] = jaz[q];
  }
  __syncthreads();
  for (int idx = tid; idx < 16 * 16 * 3; idx += 128) {
    int bb = idx / 48, r = idx % 48, j = r / 3, c = r % 3;
    float s = jred[0][c][bb * 16 + j] + jred[1][c][bb * 16 + j] +
              jred[2][c][bb * 16 + j] + jred[3][c][bb * 16 + j];
    out[(size_t)(b0 + bb) * 63 + (size_t)j * 3 + c] = s;
  }
}

// ---------------------------------------------------------------------------
extern "C" void kernel_launch(void* const* d_in, const int* in_sizes, int n_in,
                              void* d_out, int out_size, void* d_ws, size_t ws_size,
                              hipStream_t stream)
{
  const float* beta       = (const float*)d_in[0];
  const float* theta      = (const float*)d_in[1];
  const float* shapedirs  = (const float*)d_in[2];
  const float* posedirs   = (const float*)d_in[3];
  const float* v_template = (const float*)d_in[4];
  const float* Jreg       = (const float*)d_in[5];
  const float* weights    = (const float*)d_in[6];
  const float* hands_mean = (const float*)d_in[7];
  const float* hc         = (const float*)d_in[8];
  float* out = (float*)d_out;

  float* ws   = (float*)d_ws;
  float* SJ   = ws;                       // 480
  float* Jt   = ws + 480;                 // 48
  float* pf   = ws + 528;                 // BATCH * 136
  float* Amat = pf + (size_t)BATCH * NPOSEP;  // BATCH * 256

  mano_precompute<<<3, 256, 0, stream>>>(shapedirs, v_template, Jreg, SJ, Jt);
  mano_pose<<<BATCH / 256, 256, 0, stream>>>(beta, theta, hands_mean, hc, SJ, Jt, pf, Amat);
  mano_lbs<<<BATCH / 16, 128, 0, stream>>>(posedirs, shapedirs, v_template, weights,
                                           Jreg, beta, pf, Amat, out);
}